// MSDeformAttn_89687507075619
// MI455X (gfx1250) — compile-verified
//
#include <hip/hip_runtime.h>
#include <math.h>

// ---------------- problem constants ----------------
#define D_MODEL   256
#define N_HEADS   8
#define N_LEVELS  4
#define N_POINTS  4
#define LEN_V     17821
#define BS        16
#define LEN_Q     300
#define D_HEAD    32
#define K_DIM     256     // all GEMMs have K = 256

// level geometry
__device__ __constant__ int c_LH[4]    = {100, 50, 25, 13};
__device__ __constant__ int c_LW[4]    = {134, 67, 34, 17};
__device__ __constant__ int c_LSTART[4]= {0, 13400, 16750, 17600};

// ---------------- WMMA types ----------------
typedef __attribute__((ext_vector_type(2))) float v2f;
typedef __attribute__((ext_vector_type(8))) float v8f;

// =====================================================================
// GEMM:  C[M,N] = A[M,K=256] @ W[N,K=256]^T + bias[N]
// fp32 WMMA 16x16x4. Block = 256 threads = 8 waves.
// Each block: 8 M-tiles (16 rows each) x 32 N-cols; W strip staged in LDS.
// =====================================================================
#define NSTRIP     32
#define LDS_STRIDE 260   // 256 + 4 floats pad: keeps rows 16B-aligned, spreads banks

__global__ __launch_bounds__(256) void wmma_gemm_bias(
    const float* __restrict__ A, const float* __restrict__ Wt,
    const float* __restrict__ bias, float* __restrict__ C,
    int Mtiles, int N)
{
    __shared__ float lw[NSTRIP * LDS_STRIDE];
    const int n0 = blockIdx.y * NSTRIP;

    // cooperative, coalesced float4 load of the 32x256 W strip into LDS
    for (int i = threadIdx.x; i < NSTRIP * (K_DIM / 4); i += 256) {
        const int r  = i >> 6;     // 64 float4 per row
        const int c4 = i & 63;
        const float4 w4 = *reinterpret_cast<const float4*>(
            Wt + (size_t)(n0 + r) * K_DIM + (size_t)c4 * 4);
        *reinterpret_cast<float4*>(&lw[r * LDS_STRIDE + c4 * 4]) = w4;
    }
    __syncthreads();

    const int wave   = threadIdx.x >> 5;
    const int lane   = threadIdx.x & 31;
    const int m_tile = blockIdx.x * 8 + wave;
    if (m_tile >= Mtiles) return;           // wave-uniform: EXEC all-1s below

    const int lrow  = lane & 15;            // M row (A) / N col (B) within tile
    const int khalf = (lane >> 4) << 1;     // 0 for lanes 0-15, 2 for lanes 16-31

    const float* aptr = A + (size_t)(m_tile * 16 + lrow) * K_DIM + khalf;
    const float* b0p  = &lw[ lrow       * LDS_STRIDE + khalf];
    const float* b1p  = &lw[(16 + lrow) * LDS_STRIDE + khalf];

    v8f acc0 = {};
    v8f acc1 = {};
#pragma unroll 8
    for (int k0 = 0; k0 < K_DIM; k0 += 4) {
        const v2f a  = *reinterpret_cast<const v2f*>(aptr + k0);
        const v2f b0 = *reinterpret_cast<const v2f*>(b0p + k0);
        const v2f b1 = *reinterpret_cast<const v2f*>(b1p + k0);
        acc0 = __builtin_amdgcn_wmma_f32_16x16x4_f32(
                   false, a, false, b0, (short)0, acc0, false, false);
        acc1 = __builtin_amdgcn_wmma_f32_16x16x4_f32(
                   false, a, false, b1, (short)0, acc1, false, false);
    }

    // C/D layout: VGPR v holds row (8*(lane>>4) + v), col (lane&15)
    const int   crow = m_tile * 16 + ((lane >> 4) << 3);
    const float bb0  = bias[n0 + lrow];
    const float bb1  = bias[n0 + 16 + lrow];
    float* cp0 = C + (size_t)crow * N + n0 + lrow;
    float* cp1 = cp0 + 16;
#pragma unroll
    for (int v = 0; v < 8; ++v) {
        cp0[(size_t)v * N] = acc0[v] + bb0;
        cp1[(size_t)v * N] = acc1[v] + bb1;
    }
}

// =====================================================================
// Softmax over the 16 (level,point) logits per head + sampling locations
// One block per (b,q); 128 threads, thread t = h*16 + l*4 + p
// =====================================================================
__global__ __launch_bounds__(128) void softmax_loc_kernel(
    const float* __restrict__ off_raw,   // [BQ, 256]
    const float* __restrict__ attn_raw,  // [BQ, 128]
    const float* __restrict__ refer,     // [BQ, 4, 2]
    float* __restrict__ loc,             // [BQ, 128, 2]
    float* __restrict__ aw_sm)           // [BQ, 128]
{
    const int bq = blockIdx.x;
    const int t  = threadIdx.x;          // 0..127
    const int h  = t >> 4;
    const int l  = (t >> 2) & 3;

    __shared__ float sm[128];
    const float a = attn_raw[(size_t)bq * 128 + t];
    sm[t] = a;
    __syncthreads();

    const int g = h * 16;
    float mx = sm[g];
#pragma unroll
    for (int j = 1; j < 16; ++j) mx = fmaxf(mx, sm[g + j]);
    float s = 0.f;
#pragma unroll
    for (int j = 0; j < 16; ++j) s += __expf(sm[g + j] - mx);
    aw_sm[(size_t)bq * 128 + t] = __expf(a - mx) / s;

    // locations: loc = refer_bbox[b,q,l,:] + off / (W_l, H_l)
    const float ox = off_raw[(size_t)bq * 256 + 2 * t + 0];
    const float oy = off_raw[(size_t)bq * 256 + 2 * t + 1];
    const float rx = refer[(size_t)bq * 8 + l * 2 + 0];
    const float ry = refer[(size_t)bq * 8 + l * 2 + 1];
    loc[(size_t)bq * 256 + 2 * t + 0] = rx + ox / (float)c_LW[l];
    loc[(size_t)bq * 256 + 2 * t + 1] = ry + oy / (float)c_LH[l];
}

// =====================================================================
// Bilinear sampling + attention-weighted sum.
// One block per (b,q); 256 threads; thread t = h*32 + d.
// Each wave = one head: the 32 lanes read 32 consecutive floats of v
// per corner -> every gather is one coalesced 128B line.
// =====================================================================
__global__ __launch_bounds__(256) void sample_kernel(
    const float* __restrict__ v,      // [BS*LEN_V, 256] (head-major channels)
    const float* __restrict__ loc,    // [BQ, 128, 2]
    const float* __restrict__ aw,     // [BQ, 128]
    float* __restrict__ samp)         // [BQ, 256]
{
    const int bq = blockIdx.x;
    const int b  = bq / LEN_Q;
    const int t  = threadIdx.x;
    const int h  = t >> 5;
    const int d  = t & 31;

    const size_t vbase = (size_t)b * LEN_V;
    float acc = 0.f;

#pragma unroll
    for (int l = 0; l < N_LEVELS; ++l) {
        const int H = c_LH[l], W = c_LW[l], st = c_LSTART[l];
#pragma unroll
        for (int p = 0; p < N_POINTS; ++p) {
            const int i16 = h * 16 + l * 4 + p;
            const float lx = loc[(size_t)bq * 256 + 2 * i16 + 0];
            const float ly = loc[(size_t)bq * 256 + 2 * i16 + 1];
            const float w  = aw[(size_t)bq * 128 + i16];

            const float px = lx * (float)W - 0.5f;
            const float py = ly * (float)H - 0.5f;
            const float x0f = floorf(px), y0f = floorf(py);
            const float fx = px - x0f,   fy = py - y0f;
            const int   x0 = (int)x0f,   y0 = (int)y0f;

            const float cw[4] = { (1.f - fx) * (1.f - fy), fx * (1.f - fy),
                                  (1.f - fx) * fy,         fx * fy };
            const int cx[4] = { x0, x0 + 1, x0,     x0 + 1 };
            const int cy[4] = { y0, y0,     y0 + 1, y0 + 1 };
#pragma unroll
            for (int c = 0; c < 4; ++c) {
                const int xi = cx[c], yi = cy[c];
                const bool inb = (xi >= 0) && (xi <= W - 1) &&
                                 (yi >= 0) && (yi <= H - 1);
                const int xc = min(max(xi, 0), W - 1);
                const int yc = min(max(yi, 0), H - 1);
                const float g = v[(vbase + (size_t)(st + yc * W + xc)) * 256
                                  + h * 32 + d];
                acc += w * (inb ? cw[c] : 0.f) * g;
            }
        }
    }
    samp[(size_t)bq * 256 + t] = acc;
}

// =====================================================================
// launch
// =====================================================================
extern "C" void kernel_launch(void* const* d_in, const int* in_sizes, int n_in,
                              void* d_out, int out_size, void* d_ws, size_t ws_size,
                              hipStream_t stream) {
    const float* query  = (const float*)d_in[0];
    const float* refer  = (const float*)d_in[1];
    const float* value  = (const float*)d_in[2];
    const float* W_off  = (const float*)d_in[3];
    const float* b_off  = (const float*)d_in[4];
    const float* W_attn = (const float*)d_in[5];
    const float* b_attn = (const float*)d_in[6];
    const float* W_val  = (const float*)d_in[7];
    const float* b_val  = (const float*)d_in[8];
    const float* W_out  = (const float*)d_in[9];
    const float* b_out  = (const float*)d_in[10];
    float* out = (float*)d_out;

    // workspace carve-out (256B aligned)
    char* ws = (char*)d_ws;
    auto carve = [&](size_t bytes) {
        char* p = ws;
        ws += (bytes + 255) & ~(size_t)255;
        return (float*)p;
    };
    const int BQ = BS * LEN_Q;                               // 4800
    float* v_ws  = carve((size_t)BS * LEN_V * D_MODEL * 4);  // 292 MB
    float* offr  = carve((size_t)BQ * 256 * 4);
    float* attnr = carve((size_t)BQ * 128 * 4);
    float* locb  = carve((size_t)BQ * 256 * 4);
    float* awsm  = carve((size_t)BQ * 128 * 4);
    float* samp  = carve((size_t)BQ * 256 * 4);

    // 1) value projection: [BS*LEN_V,256] @ W_val^T + b_val
    {
        const int Mtiles = (BS * LEN_V) / 16;                // 17821
        dim3 grid((Mtiles + 7) / 8, D_MODEL / NSTRIP);
        wmma_gemm_bias<<<grid, 256, 0, stream>>>(value, W_val, b_val, v_ws,
                                                 Mtiles, D_MODEL);
    }
    // 2) offset projection: [4800,256] @ W_off^T + b_off
    {
        const int Mtiles = BQ / 16;                          // 300
        dim3 grid((Mtiles + 7) / 8, 256 / NSTRIP);
        wmma_gemm_bias<<<grid, 256, 0, stream>>>(query, W_off, b_off, offr,
                                                 Mtiles, 256);
    }
    // 3) attention projection: [4800,256] @ W_attn^T + b_attn
    {
        const int Mtiles = BQ / 16;
        dim3 grid((Mtiles + 7) / 8, 128 / NSTRIP);
        wmma_gemm_bias<<<grid, 256, 0, stream>>>(query, W_attn, b_attn, attnr,
                                                 Mtiles, 128);
    }
    // 4) softmax + sampling locations
    softmax_loc_kernel<<<BQ, 128, 0, stream>>>(offr, attnr, refer, locb, awsm);
    // 5) bilinear sampling + weighted sum
    sample_kernel<<<BQ, 256, 0, stream>>>(v_ws, locb, awsm, samp);
    // 6) output projection: [4800,256] @ W_out^T + b_out -> d_out
    {
        const int Mtiles = BQ / 16;
        dim3 grid((Mtiles + 7) / 8, 256 / NSTRIP);
        wmma_gemm_bias<<<grid, 256, 0, stream>>>(samp, W_out, b_out, out,
                                                 Mtiles, 256);
    }
}